// TreeRecursiveTreeOnlyNN_77790447665825
// MI455X (gfx1250) — compile-verified
//
#include <hip/hip_runtime.h>
#include <hip/hip_bf16.h>
#include <math.h>

// ---------------------------------------------------------------------------
// Uniform-tree Tree-LSTM.
//
// All leaves start from identical zero (h,c); the level recurrence maps
// identical children to identical parents, so every level has ONE distinct
// (h,c) state. We iterate L single-node steps:
//     z[320] = (Wl + Wr) . h          (fp32 WMMA 16x16x4)
//     f_l,f_r,i,o = sigmoid(z[0..3]), u = tanh(z[4])
//     c' = i*u + (f_l+f_r)*c ;  h' = o*tanh(c')
// Root: root_hidden = concat(h, h); scores = Ws @ root_hidden + bs.
// Output: d_out[0:3] = scores, d_out[3:131] = root_hidden.
// ---------------------------------------------------------------------------

typedef __attribute__((ext_vector_type(2))) float v2f;
typedef __attribute__((ext_vector_type(8))) float v8f;

#define HDIM   64
#define GATES  5
#define ZOUT   (GATES * HDIM)   // 320 gate pre-activations
#define NWAVES (ZOUT / 16)      // 20 waves, one 16-wide output tile each
#define KSTEPS (HDIM / 4)       // 16 WMMA K-steps of 4

__global__ __launch_bounds__(NWAVES * 32)
void tree_lstm_uniform_kernel(const float* __restrict__ Wl,
                              const float* __restrict__ Wr,
                              const float* __restrict__ Ws,
                              const float* __restrict__ bs,
                              const int*   __restrict__ Lp,
                              float* __restrict__ out)
{
    __shared__ float sh[HDIM];   // current hidden state (uniform across nodes)
    __shared__ float sc[HDIM];   // current cell state
    __shared__ float sz[ZOUT];   // gate pre-activations

    const int tid   = threadIdx.x;
    const int lane  = tid & 31;
    const int wave  = tid >> 5;
    const int half  = lane >> 4;      // 0: lanes 0-15 (K=0,1), 1: lanes 16-31 (K=2,3)
    const int col   = lane & 15;      // N index inside the 16x16 tile
    const int obase = wave * 16;      // this wave's output-row base in z

    // ---- Load B fragments once: W = Wl + Wr, tile (K=4 x N=16) per step. ----
    // B[k][n] = W[obase+n][kb+k]; per ISA 32-bit operand layout, VGPR0 holds
    // K=0 (lanes 0-15) / K=2 (lanes 16-31), VGPR1 holds K=1 / K=3.
    v2f bfrag[KSTEPS];
    {
        const int O = obase + col;                 // output row (gate*64 + o)
        const float* wl = Wl + O * HDIM;
        const float* wr = Wr + O * HDIM;
#pragma unroll
        for (int s = 0; s < KSTEPS; ++s) {
            const int k0 = 4 * s + 2 * half;
            bfrag[s].x = wl[k0]     + wr[k0];
            bfrag[s].y = wl[k0 + 1] + wr[k0 + 1];
        }
    }

    // ---- init h = c = 0 (leaf children are init_hidden() zeros) ----
    for (int i = tid; i < HDIM; i += blockDim.x) { sh[i] = 0.0f; sc[i] = 0.0f; }
    __syncthreads();

    const int L = Lp[0];

    for (int lvl = 0; lvl < L; ++lvl) {
        // A operand: h replicated across all 16 M rows -> every D row equals z.
        // A[m][k] = h[kb+k] for all m; same lane-half K split as B.
        v8f acc = {};
#pragma unroll
        for (int s = 0; s < KSTEPS; ++s) {
            const int k0 = 4 * s + 2 * half;
            v2f a;
            a.x = sh[k0];
            a.y = sh[k0 + 1];
            // D = A x B + C  (fp32, 16x16x4) — accumulator-chained over K.
            acc = __builtin_amdgcn_wmma_f32_16x16x4_f32(
                      /*neg_a=*/false, a,
                      /*neg_b=*/false, bfrag[s],
                      /*c_mod=*/(short)0, acc,
                      /*reuse_a=*/false, /*reuse_b=*/false);
        }
        // D[m][n] = z[obase+n] for every m; lane n (0-15) reads VGPR0.
        if (half == 0) sz[obase + col] = acc[0];
        __syncthreads();

        // Gate math on 64 lanes; gate order: [f_l, f_r, i, o, u].
        if (tid < HDIM) {
            const float fl = 1.0f / (1.0f + expf(-sz[tid]));
            const float fr = 1.0f / (1.0f + expf(-sz[HDIM + tid]));
            const float ig = 1.0f / (1.0f + expf(-sz[2 * HDIM + tid]));
            const float og = 1.0f / (1.0f + expf(-sz[3 * HDIM + tid]));
            const float u  = tanhf(sz[4 * HDIM + tid]);
            const float cn = ig * u + (fl + fr) * sc[tid];  // c_l == c_r == c
            sc[tid] = cn;
            sh[tid] = og * tanhf(cn);
        }
        __syncthreads();
    }

    // ---- Root: root_hidden = concat(h[0], h[1]) = [h, h] (uniform level). ----
    if (tid < 2 * HDIM) out[3 + tid] = sh[tid & (HDIM - 1)];
    if (tid < 3) {
        float s = bs[tid];
        const float* wsr = Ws + tid * (2 * HDIM);
        for (int k = 0; k < HDIM; ++k)
            s += sh[k] * (wsr[k] + wsr[HDIM + k]);
        out[tid] = s;
    }
}

extern "C" void kernel_launch(void* const* d_in, const int* in_sizes, int n_in,
                              void* d_out, int out_size, void* d_ws, size_t ws_size,
                              hipStream_t stream) {
    const float* Wl = (const float*)d_in[0];   // [5,64,64]
    const float* Wr = (const float*)d_in[1];   // [5,64,64]
    const float* Ws = (const float*)d_in[2];   // [3,128]
    const float* bs = (const float*)d_in[3];   // [3]
    const int*   L  = (const int*)d_in[4];     // scalar level count
    float* out = (float*)d_out;                // [131] = 3 scores + 128 root

    tree_lstm_uniform_kernel<<<1, NWAVES * 32, 0, stream>>>(Wl, Wr, Ws, bs, L, out);
}